// SimpleUpscaleConv2d_23029614641179
// MI455X (gfx1250) — compile-verified
//
#include <hip/hip_runtime.h>

typedef __attribute__((ext_vector_type(2))) float v2f;
typedef __attribute__((ext_vector_type(8))) float v8f;
typedef __attribute__((ext_vector_type(4))) unsigned int v4u;
typedef __attribute__((ext_vector_type(8))) int v8i;
typedef __attribute__((ext_vector_type(4))) int v4i;

#define HH   512
#define WW   512
#define CIN  16
#define COUT 16
#define NB   8
#define RT   8           // input rows per tile
#define JT   64          // input cols per tile
#define LROWS 10         // RT + 2 halo
#define LCOLS 66         // JT + 2 halo (= TDM tile_dim0 = LDS pitch, contiguous)
#define LPITCH 66
#define LCH  (LROWS * LPITCH)   // 660 floats per channel
#define WSCALE 0.11785113019775793f  // sqrt(2)/sqrt(16*3*3)

__global__ __launch_bounds__(256) void upconv2x_wmma(
    const float* __restrict__ x,      // [8,16,512,512]
    const float* __restrict__ wgt,    // [16,16,3,3]
    const float* __restrict__ bias,   // [16]
    float* __restrict__ out)          // [8,16,1024,1024]
{
    __shared__ float swgt[COUT * CIN * 9];
    __shared__ float sbias[16];
    __shared__ float sin[CIN * LCH + 80];   // +80 guard for TDM edge-spill

    const int tid  = threadIdx.x;
    const int lane = tid & 31;
    const int wv   = tid >> 5;

    // tile decode: 8 col-tiles x 64 row-tiles x 8 batches = 4096 blocks
    const int t  = blockIdx.x;
    const int jb = t & 7;
    const int ib = (t >> 3) & 63;
    const int n  = t >> 9;
    const int i0 = ib * RT;
    const int j0 = jb * JT;

    // ---- stage weights + bias into LDS (small; plain loads) ----
    for (int k = tid; k < COUT * CIN * 9; k += 256) swgt[k] = wgt[k];
    if (tid < 16) sbias[tid] = bias[tid];

    // ---- stage the whole input tile (16ch x 10 x 66 halo tile) with ONE TDM op ----
    // Clamp top/left start (negative coords not expressible); shift LDS placement;
    // right/bottom OOB returns zero from TDM via tensor_dim checks.
    if (wv == 0) {
        int rstart = i0 - 1, lr0 = 0;
        if (rstart < 0) { rstart = 0; lr0 = 1; }
        int cstart = j0 - 1, lc0 = 0;
        if (cstart < 0) { cstart = 0; lc0 = 1; }

        unsigned lds_off = (unsigned)(size_t)(&sin[0]) + 4u * (unsigned)(lr0 * LPITCH + lc0);
        unsigned long long ga = (unsigned long long)(const void*)
            (x + (size_t)n * CIN * HH * WW + (size_t)rstart * WW + cstart);

        const unsigned td0 = (unsigned)(WW - cstart);   // tensor_dim0 (cols remaining)
        const unsigned td1 = (unsigned)(HH - rstart);   // tensor_dim1 (rows remaining)

        v4u g0;
        g0[0] = 1u;                                     // count=1, user mode
        g0[1] = lds_off;                                // lds_addr [63:32]
        g0[2] = (unsigned)(ga & 0xFFFFFFFFu);           // global_addr lo
        g0[3] = (unsigned)(((ga >> 32) & 0x1FFFFFFu) | (2u << 30)); // addr hi | type=2

        v8i g1;
        g1[0] = (int)(2u << 16);                        // data_size=2 (4 bytes); mask=0; no pad
        g1[1] = (int)((td0 & 0xFFFFu) << 16);           // tensor_dim0[15:0] @ [63:48]
        g1[2] = (int)(((td0 >> 16) & 0xFFFFu) | ((td1 & 0xFFFFu) << 16));
        g1[3] = (int)(((td1 >> 16) & 0xFFFFu) | ((unsigned)LCOLS << 16)); // tile_dim0=66
        g1[4] = (int)((unsigned)LROWS | ((unsigned)CIN << 16));           // tile_dim1=10, tile_dim2=16
        g1[5] = (int)(unsigned)WW;                      // tensor_dim0_stride lo32 = 512
        g1[6] = (int)(((unsigned)(HH * WW) & 0xFFFFu) << 16); // stride0 hi16=0 | stride1[15:0]
        g1[7] = (int)((unsigned)(HH * WW) >> 16);       // tensor_dim1_stride hi = 262144>>16

        v4i g2;
        g2[0] = CIN;                                    // tensor_dim2 = 16
        g2[1] = 0;                                      // tensor_dim3 (unused)
        g2[2] = 0;                                      // tensor_dim2_stride lo (unused, 3D)
        g2[3] = 0;                                      // stride hi | tile_dim3=0 (unused)
        v4i g3;
        g3[0] = 0; g3[1] = 0; g3[2] = 0; g3[3] = 0;     // dims 4+ unused

        v8i g4;                                         // extra arg on 6-arg toolchain (zero-filled)
        g4[0] = 0; g4[1] = 0; g4[2] = 0; g4[3] = 0;
        g4[4] = 0; g4[5] = 0; g4[6] = 0; g4[7] = 0;

        __builtin_amdgcn_tensor_load_to_lds(g0, g1, g2, g3, g4, 0);
        __builtin_amdgcn_s_wait_tensorcnt(0);
    }
    __syncthreads();

    // ---- zero-fill top/left halo (and TDM spill cells) for edge blocks ----
    if (i0 == 0) {
        for (int k = tid; k < CIN * LCOLS; k += 256)
            sin[(k / LCOLS) * LCH + (k % LCOLS)] = 0.0f;          // row 0, all channels
    }
    if (j0 == 0) {
        for (int k = tid; k < CIN * LROWS; k += 256)
            sin[(k / LROWS) * LCH + (k % LROWS) * LPITCH] = 0.0f; // col 0, all rows/channels
    }
    __syncthreads();

    // ---- per-lane A fragments (effective 2x2 kernels per parity) ----
    const int o  = lane & 15;   // out channel (M row of A)
    const int aL = lane >> 4;   // 0: K rows {0,1}=a0 ; 1: K rows {2,3}=a1
    const int py = wv >> 2;     // waves 0-3: even output rows, 4-7: odd

    v2f A0[16], A1[16];         // px=0 and px=1 fragments, one per channel-chunk
    #pragma unroll
    for (int c = 0; c < 16; ++c) {
        const float* wp = &swgt[(o * 16 + c) * 9];
        float wr0, wr1, wr2;
        {
            float a0 = wp[0], a1 = wp[1], a2 = wp[2];   // ky=0 row
            float b0 = wp[3], b1 = wp[4], b2 = wp[5];   // ky=1 row
            float c0 = wp[6], c1 = wp[7], c2 = wp[8];   // ky=2 row
            if (py == 0) {
                wr0 = (aL == 0) ? a0 : (b0 + c0);
                wr1 = (aL == 0) ? a1 : (b1 + c1);
                wr2 = (aL == 0) ? a2 : (b2 + c2);
            } else {
                wr0 = (aL == 0) ? (a0 + b0) : c0;
                wr1 = (aL == 0) ? (a1 + b1) : c1;
                wr2 = (aL == 0) ? (a2 + b2) : c2;
            }
        }
        A0[c].x = wr0 * WSCALE;          // px=0, b=0  (col j-1)
        A0[c].y = (wr1 + wr2) * WSCALE;  // px=0, b=1  (col j)
        A1[c].x = (wr0 + wr1) * WSCALE;  // px=1, b=0  (col j)
        A1[c].y = wr2 * WSCALE;          // px=1, b=1  (col j+1)
    }

    // accumulator init = bias broadcast (D layout: lane<16 -> M=v, lane>=16 -> M=v+8)
    v8f cinit;
    #pragma unroll
    for (int v = 0; v < 8; ++v) cinit[v] = sbias[v + 8 * aL];

    // ---- main compute ----
    const int iw   = wv & 3;
    const int ncol = lane & 15;
    float* outb = out + (size_t)n * COUT * 1024 * 1024;

    #pragma unroll
    for (int tt = 0; tt < 2; ++tt) {
        const int iLocal = iw + 4 * tt;          // 2 input rows per wave
        const int r0 = iLocal + py;              // LDS row base (global row i-1+py)
        const int Y  = 2 * (i0 + iLocal) + py;   // output row
        const float* rowp = &sin[(r0 + aL) * LPITCH + ncol];

        for (int jw = 0; jw < JT; jw += 16) {
            v8f d0 = cinit;   // px = 0 tile
            v8f d1 = cinit;   // px = 1 tile
            #pragma unroll
            for (int c = 0; c < 16; ++c) {
                const float* p = rowp + c * LCH + jw;
                float vL = p[0];   // col j-1 (+n)
                float vM = p[1];   // col j   (+n)
                float vR = p[2];   // col j+1 (+n)
                v2f b0; b0.x = vL; b0.y = vM;   // B for px=0 (K{0,2}/K{1,3})
                v2f b1; b1.x = vM; b1.y = vR;   // B for px=1
                d0 = __builtin_amdgcn_wmma_f32_16x16x4_f32(
                        false, A0[c], false, b0, (short)0, d0, false, false);
                d1 = __builtin_amdgcn_wmma_f32_16x16x4_f32(
                        false, A1[c], false, b1, (short)0, d1, false, false);
            }
            // interleave px0/px1 -> coalesced float2 stores (32 consecutive X per M row)
            const int X0 = 2 * (j0 + jw + ncol);
            #pragma unroll
            for (int v = 0; v < 8; ++v) {
                const int oo = v + 8 * aL;
                v2f pr; pr.x = d0[v]; pr.y = d1[v];
                *(v2f*)&outb[(size_t)oo * (1024 * 1024) + (size_t)Y * 1024 + X0] = pr;
            }
        }
    }
}

extern "C" void kernel_launch(void* const* d_in, const int* in_sizes, int n_in,
                              void* d_out, int out_size, void* d_ws, size_t ws_size,
                              hipStream_t stream) {
    const float* x    = (const float*)d_in[0];
    const float* wgt  = (const float*)d_in[1];
    const float* bias = (const float*)d_in[2];
    float* out = (float*)d_out;

    const int nblocks = NB * (HH / RT) * (WW / JT);  // 8 * 64 * 8 = 4096
    upconv2x_wmma<<<nblocks, 256, 0, stream>>>(x, wgt, bias, out);
}